// HierarchicalNLLWithAnalyticalGradient_64707977281689
// MI455X (gfx1250) — compile-verified
//
#include <hip/hip_runtime.h>
#include <stddef.h>

typedef __attribute__((ext_vector_type(2))) float v2f;
typedef __attribute__((ext_vector_type(8))) float v8f;

#define K_STATES 12
#define N_B 64
#define N_T 400
#define N_D 80
#define LSTR 84                 // padded LDS row stride (dwords), even for b64 loads
#define LOG2PI 1.8378770664093453f

// One block per (b,k). 256 threads = 8 wave32.
__global__ __launch_bounds__(256)
void hnll_kernel(const float* __restrict__ data,      // (B,T,D)
                 const float* __restrict__ mu_subj,   // (M,K,D)
                 const float* __restrict__ L_subj,    // (M,K,D,D)
                 const float* __restrict__ gamma_,    // (B,T,K)
                 const int*   __restrict__ sids,      // (B,)
                 float* __restrict__ partials,        // (B*K,) in d_ws
                 float* __restrict__ out,             // scalar (atomic fallback)
                 int use_atomic)
{
  __shared__ __align__(16) float Ls[N_D * N_D];      // original L, row-major
  __shared__ __align__(16) float Xi[N_D * LSTR];     // L^{-1}, padded rows
  __shared__ __align__(16) float smu[N_D];
  __shared__ float sinv[N_D];                        // 1 / diag(L)
  __shared__ float lg[N_D];                          // log(diag(L))
  __shared__ float s_logdet;
  __shared__ float s_red[8];

  const int blk = blockIdx.x;
  const int b   = blk / K_STATES;
  const int k   = blk - b * K_STATES;
  const int tid = threadIdx.x;
  const int sid = sids[b];

  const float* Lg  = L_subj  + (size_t)(sid * K_STATES + k) * (N_D * N_D);
  const float* mug = mu_subj + (size_t)(sid * K_STATES + k) * N_D;

  // ---- stage L and mu into LDS; zero Xi ----
  for (int i = tid; i < (N_D * N_D) / 4; i += 256)
    ((float4*)Ls)[i] = ((const float4*)Lg)[i];
  for (int i = tid; i < N_D; i += 256) smu[i] = mug[i];
  for (int i = tid; i < N_D * LSTR; i += 256) Xi[i] = 0.0f;
  __syncthreads();
  if (tid < N_D) {
    const float d = Ls[tid * N_D + tid];
    Xi[tid * LSTR + tid] = 1.0f;
    sinv[tid] = 1.0f / d;
    lg[tid] = logf(d);
  }
  __syncthreads();

  // ---- Xi <- unscaled L^{-1} by single-barrier-per-pivot sweep ----
  // Keep rows unscaled: U[p] = e_p - sum_{q<p} L[p][q] * (inv_q * U[q]);
  // eliminate with factor f = L[i][p] * inv_p; scale all rows at the end.
  // L entries read from original Ls are valid throughout (pivot rows of the
  // L-part reduce to e_p before they are ever reused).
  for (int p = 0; p < N_D; ++p) {
    const float invp = sinv[p];
    const int ncols = p + 1;                  // U[p] has nonzeros in cols <= p
    const int nel = (N_D - 1 - p) * ncols;
    for (int e = tid; e < nel; e += 256) {
      const int r = e / ncols;
      const int j = e - r * ncols;
      const int i = p + 1 + r;
      Xi[i * LSTR + j] -= (Ls[i * N_D + p] * invp) * Xi[p * LSTR + j];
    }
    __syncthreads();
  }
  // final scaling: X[i][j] = inv_i * U[i][j]  (lower triangle only)
  for (int e = tid; e < N_D * N_D; e += 256) {
    const int i = e / N_D;
    const int j = e - i * N_D;
    if (j <= i) Xi[i * LSTR + j] *= sinv[i];
  }
  if (tid == 0) {                      // deterministic fixed-order logdet
    float s = 0.0f;
    for (int i = 0; i < N_D; ++i) s += lg[i];
    s_logdet = s;
  }
  __syncthreads();

  const float logdet = s_logdet;
  const float cterm  = -0.5f * (float)N_D * LOG2PI;

  const int wave  = tid >> 5;
  const int lane  = tid & 31;
  const int nloc  = lane & 15;   // N (t) column within tile / M row for A-frag
  const int khalf = lane >> 4;   // selects K pair {0,1} vs {2,3} in a k-step

  float local = 0.0f;
  // 25 t-tiles of 16 columns, distributed over 8 waves (wave-uniform loop,
  // EXEC stays all-ones for the WMMAs).
  for (int tt = wave; tt < N_T / 16; tt += 8) {
    const int t = tt * 16 + nloc;
    const float* dp = data + ((size_t)b * N_T + t) * N_D;

    // B fragments: diff[d][t] = data[b][t][d] - mu[d]; 4x16 tile per k-step.
    // Lane layout: VGPR0 = B[2*khalf][n], VGPR1 = B[2*khalf+1][n].
    v2f bf[20];
#pragma unroll
    for (int s = 0; s < 20; ++s) {
      const int d0 = 4 * s + 2 * khalf;          // even -> 8B aligned
      v2f dv = *(const v2f*)(dp + d0);
      v2f mv = *(const v2f*)(smu + d0);
      bf[s] = dv - mv;
    }

    float quad = 0.0f;
    // m-loop kept rolled: A-fragments are re-read from LDS each m-tile
    // (ds pipe co-executes with WMMA), keeping VGPR pressure < 256 and
    // avoiding s_set_vgpr_msb churn / low occupancy.
#pragma unroll 1
    for (int m = 0; m < 5; ++m) {
      v8f c = {};
      const float* arow = &Xi[(m * 16 + nloc) * LSTR];
#pragma unroll
      for (int s = 0; s < 20; ++s) {
        const int d0 = 4 * s + 2 * khalf;
        // A frag (16x4 f32): lane M = nloc, VGPR0/1 = K = d0, d0+1
        v2f af = *(const v2f*)(arow + d0);
        c = __builtin_amdgcn_wmma_f32_16x16x4_f32(
                false, af, false, bf[s], (short)0, c, false, false);
      }
      // C layout: lane n = nloc, VGPR v holds row m16 = v + 8*khalf
#pragma unroll
      for (int v = 0; v < 8; ++v) quad += c[v] * c[v];
    }
    // combine row-halves (lanes l and l+16 hold same column t)
    quad += __shfl_xor(quad, 16, 32);
    if (lane < 16) {
      const float lp = -0.5f * quad - logdet + cterm;
      local += gamma_[((size_t)b * N_T + t) * K_STATES + k] * lp;
    }
  }

  // fixed-pattern wave + block reduction (deterministic)
  for (int off = 16; off > 0; off >>= 1) local += __shfl_xor(local, off, 32);
  if (lane == 0) s_red[wave] = local;
  __syncthreads();
  if (tid == 0) {
    float s = 0.0f;
    for (int w = 0; w < 8; ++w) s += s_red[w];
    if (use_atomic) atomicAdd(out, -s * (1.0f / ((float)N_B * (float)N_T)));
    else            partials[blk] = s;
  }
}

__global__ void hnll_zero_out(float* out) {
  if (threadIdx.x == 0) out[0] = 0.0f;
}

__global__ __launch_bounds__(256)
void hnll_finalize(const float* __restrict__ partials, float* __restrict__ out, int n) {
  __shared__ float sm[256];
  const int tid = threadIdx.x;
  float s = 0.0f;
  for (int i = tid; i < n; i += 256) s += partials[i];   // fixed order
  sm[tid] = s;
  __syncthreads();
  for (int o = 128; o > 0; o >>= 1) {
    if (tid < o) sm[tid] += sm[tid + o];
    __syncthreads();
  }
  if (tid == 0) out[0] = -sm[0] * (1.0f / ((float)N_B * (float)N_T));
}

extern "C" void kernel_launch(void* const* d_in, const int* in_sizes, int n_in,
                              void* d_out, int out_size, void* d_ws, size_t ws_size,
                              hipStream_t stream) {
  (void)in_sizes; (void)n_in; (void)out_size;
  const float* data    = (const float*)d_in[0];
  const float* mu_subj = (const float*)d_in[1];
  const float* L_subj  = (const float*)d_in[2];
  const float* gamma_  = (const float*)d_in[3];
  const int*   sids    = (const int*)d_in[4];
  float* out = (float*)d_out;

  const int nblk = N_B * K_STATES;                 // 768
  const bool have_ws = (ws_size >= (size_t)nblk * sizeof(float));
  float* partials = (float*)d_ws;

  if (!have_ws) {
    hnll_zero_out<<<1, 1, 0, stream>>>(out);
    hnll_kernel<<<nblk, 256, 0, stream>>>(data, mu_subj, L_subj, gamma_, sids,
                                          out /*unused as partials*/, out, 1);
  } else {
    hnll_kernel<<<nblk, 256, 0, stream>>>(data, mu_subj, L_subj, gamma_, sids,
                                          partials, out, 0);
    hnll_finalize<<<1, 256, 0, stream>>>(partials, out, nblk);
  }
}